// KNNSoftmax_81621558493819
// MI455X (gfx1250) — compile-verified
//
#include <hip/hip_runtime.h>
#include <math.h>

// Problem constants (from reference): N=8192, D=128, K=16, ALPHA=1.0
#define NTOT  8192
#define DDIM  128
#define KSEL  16
#define ALPHA_F 1.0f
#define INF_BITS 0x7F800000u

// LDS B-tile: 64 rows x (128 + 4 pad) dwords  -> conflict-free b64 frag reads
#define BROWS   64
#define BSTRIDE 132

typedef float v2f __attribute__((ext_vector_type(2)));
typedef float v8f __attribute__((ext_vector_type(8)));
typedef unsigned int v4u __attribute__((ext_vector_type(4)));
typedef int v4i __attribute__((ext_vector_type(4)));
typedef int v8i __attribute__((ext_vector_type(8)));

// ---------------- block reduction helpers (wave32, 8 waves / 256 threads) ----
__device__ __forceinline__ int blockSumI(int v, int* s) {
  #pragma unroll
  for (int m = 16; m; m >>= 1) v += __shfl_xor(v, m, 32);
  __syncthreads();
  if ((threadIdx.x & 31) == 0) s[threadIdx.x >> 5] = v;
  __syncthreads();
  int r = s[0];
  #pragma unroll
  for (int k = 1; k < 8; ++k) r += s[k];
  return r;
}

__device__ __forceinline__ float blockSumF(float v, float* s) {
  #pragma unroll
  for (int m = 16; m; m >>= 1) v += __shfl_xor(v, m, 32);
  __syncthreads();
  if ((threadIdx.x & 31) == 0) s[threadIdx.x >> 5] = v;
  __syncthreads();
  float r = s[0];
  #pragma unroll
  for (int k = 1; k < 8; ++k) r += s[k];
  return r;
}

__device__ __forceinline__ int blockMinI(int v, int* s) {
  #pragma unroll
  for (int m = 16; m; m >>= 1) { int o = __shfl_xor(v, m, 32); v = o < v ? o : v; }
  __syncthreads();
  if ((threadIdx.x & 31) == 0) s[threadIdx.x >> 5] = v;
  __syncthreads();
  int r = s[0];
  #pragma unroll
  for (int k = 1; k < 8; ++k) r = s[k] < r ? s[k] : r;
  return r;
}

__device__ __forceinline__ int blockMaxI(int v, int* s) {
  #pragma unroll
  for (int m = 16; m; m >>= 1) { int o = __shfl_xor(v, m, 32); v = o > v ? o : v; }
  __syncthreads();
  if ((threadIdx.x & 31) == 0) s[threadIdx.x >> 5] = v;
  __syncthreads();
  int r = s[0];
  #pragma unroll
  for (int k = 1; k < 8; ++k) r = s[k] > r ? s[k] : r;
  return r;
}

// ---------------- K1: row squared norms; block 0 zeros the accumulators -----
__global__ void sqnorm_kernel(const float* __restrict__ X, float* __restrict__ sq,
                              float* __restrict__ accum) {
  if (blockIdx.x == 0 && threadIdx.x < 4) accum[threadIdx.x] = 0.f;
  const int wave = threadIdx.x >> 5;
  const int lane = threadIdx.x & 31;
  const int row  = blockIdx.x * 8 + wave;           // 8 rows per 256-thread block
  const float4 v = *(const float4*)(X + (size_t)row * DDIM + lane * 4);
  float s = v.x * v.x + v.y * v.y + v.z * v.z + v.w * v.w;
  #pragma unroll
  for (int m = 16; m; m >>= 1) s += __shfl_xor(s, m, 32);
  if (lane == 0) sq[row] = s;
}

// ---------------- K2: fp32 WMMA distance GEMM with TDM-staged B tile ---------
// dist[i][j] = sqrt(max(sq[i] + sq[j] - 2*dot(x_i, x_j), 1e-12))
// Block = 256 threads (8 waves, 4 row-groups x 2 col-groups) -> 128x64 output.
// B tile (64 cols x 128 k) is staged once per block into LDS via the Tensor
// Data Mover (TENSOR_LOAD_TO_LDS, TENSORcnt), with 4-dword row padding so the
// per-kk v2f fragment reads from LDS hit 64 distinct banks (conflict-free).
// A fragments stream from global (L2-resident, 4 MB total input).
__global__ void dist_kernel(const float* __restrict__ X, const float* __restrict__ sq,
                            float* __restrict__ dist) {
  __shared__ float Bs[BROWS * BSTRIDE];             // 33792 B

  const int lane = threadIdx.x & 31;
  const int wave = threadIdx.x >> 5;    // 0..7
  const int wrow = wave >> 1;           // 0..3 -> 32-row group inside 128
  const int wcol = wave & 1;            // 0..1 -> 32-col group inside 64
  const int i0  = blockIdx.y * 128 + wrow * 32;
  const int jc0 = wcol * 32;                        // LDS-relative column base
  const int j0  = blockIdx.x * 64 + jc0;
  const int half = lane >> 4;           // selects K-pair (and M group in C)
  const int r    = lane & 15;           // M for A-frags, N for B/C frags

#if __has_builtin(__builtin_amdgcn_tensor_load_to_lds)
  if (wave == 0) {
    // ---- Tensor DMA descriptor (D#) per CDNA5 ISA ch.8 ----
    const unsigned long long gaddr =
        (unsigned long long)(size_t)(X + (size_t)blockIdx.x * 64 * DDIM);
    const unsigned ldsaddr = (unsigned)(size_t)&Bs[0];
    v4u g0;
    g0[0] = 1u;                                        // count=1, user desc
    g0[1] = ldsaddr;                                   // lds_addr (bytes)
    g0[2] = (unsigned)(gaddr & 0xFFFFFFFFu);           // global_addr[31:0]
    g0[3] = (unsigned)((gaddr >> 32) & 0x1FFFFFFu)     // global_addr[56:32]
            | (2u << 30);                              // type=2 ("image")
    v8i g1;
    g1[0] = (int)(0x00020000u       // data_size=2 (4 B)
                | 0x00100000u       // pad_enable
                | (6u << 22)        // pad_interval: 128 dwords
                | (3u << 25));      // pad_amount: 4 dwords
    g1[1] = (int)(128u << 16);      // tensor_dim0 = 128 (row length, elems)
    g1[2] = (int)(8192u << 16);     // tensor_dim1 = 8192 (rows in tensor)
    g1[3] = (int)(128u << 16);      // tile_dim0 = 128 elems
    g1[4] = (int)BROWS;             // tile_dim1 = 64 rows
    g1[5] = (int)DDIM;              // tensor_dim0_stride = 128 elems
    g1[6] = 0;
    g1[7] = 0;
    v4i gz = {0, 0, 0, 0};
  #if defined(__has_include) && __has_include(<hip/amd_detail/amd_gfx1250_TDM.h>)
    v8i gz8 = {0, 0, 0, 0, 0, 0, 0, 0};
    __builtin_amdgcn_tensor_load_to_lds(g0, g1, gz, gz, gz8, 0);
  #else
    __builtin_amdgcn_tensor_load_to_lds(g0, g1, gz, gz, 0);
  #endif
    __builtin_amdgcn_s_wait_tensorcnt(0);
  }
  __syncthreads();
#else
  // Fallback: cooperative global->LDS staging with the same padded layout.
  {
    const float* Xcol = X + (size_t)blockIdx.x * 64 * DDIM;
    for (int idx = threadIdx.x; idx < BROWS * (DDIM / 4); idx += 256) {
      const int row = idx >> 5;
      const int c4  = (idx & 31) * 4;
      *(float4*)(&Bs[row * BSTRIDE + c4]) =
          *(const float4*)(Xcol + (size_t)row * DDIM + c4);
    }
    __syncthreads();
  }
#endif

  // A-frag pointers (global); B-frags come from LDS.
  const float* pa0 = X + (size_t)(i0 + r) * DDIM + 2 * half;
  const float* pa1 = pa0 + 16 * DDIM;
  const float* pb0 = Bs + (size_t)(jc0 + r) * BSTRIDE + 2 * half;
  const float* pb1 = pb0 + 16 * BSTRIDE;

  v8f c00 = {}, c01 = {}, c10 = {}, c11 = {};
  #pragma unroll 8
  for (int kk = 0; kk < DDIM / 4; ++kk) {
    v2f a0 = *(const v2f*)(pa0 + 4 * kk);
    v2f a1 = *(const v2f*)(pa1 + 4 * kk);
    v2f b0 = *(const v2f*)(pb0 + 4 * kk);
    v2f b1 = *(const v2f*)(pb1 + 4 * kk);
    c00 = __builtin_amdgcn_wmma_f32_16x16x4_f32(false, a0, false, b0, (short)0, c00, false, false);
    c01 = __builtin_amdgcn_wmma_f32_16x16x4_f32(false, a0, false, b1, (short)0, c01, false, false);
    c10 = __builtin_amdgcn_wmma_f32_16x16x4_f32(false, a1, false, b0, (short)0, c10, false, false);
    c11 = __builtin_amdgcn_wmma_f32_16x16x4_f32(false, a1, false, b1, (short)0, c11, false, false);
  }

  // C layout: VGPR v, lane l -> M = v + 8*(l>>4), N = l&15
  const float sqj0 = sq[j0 + r];
  const float sqj1 = sq[j0 + 16 + r];
  #pragma unroll
  for (int v = 0; v < 8; ++v) {
    const int iA = i0 + v + 8 * half;
    const int iB = iA + 16;
    const float sqiA = sq[iA];
    const float sqiB = sq[iB];
    dist[(size_t)iA * NTOT + j0 + r]      = sqrtf(fmaxf(sqiA + sqj0 - 2.f * c00[v], 1e-12f));
    dist[(size_t)iA * NTOT + j0 + 16 + r] = sqrtf(fmaxf(sqiA + sqj1 - 2.f * c01[v], 1e-12f));
    dist[(size_t)iB * NTOT + j0 + r]      = sqrtf(fmaxf(sqiB + sqj0 - 2.f * c10[v], 1e-12f));
    dist[(size_t)iB * NTOT + j0 + 16 + r] = sqrtf(fmaxf(sqiB + sqj1 - 2.f * c11[v], 1e-12f));
  }
}

// ---------------- K3: per-row exact select + masked reductions ---------------
// One 256-thread block per row. Each thread keeps 32 row elements in registers.
// Threshold = exact 17th smallest (diag -> +inf), via binary search on IEEE bits.
__global__ void row_kernel(const float* __restrict__ dist, const int* __restrict__ tgt,
                           float* __restrict__ accum) {
  __shared__ int   si[8];
  __shared__ float sf[8];
  const int i  = blockIdx.x;
  const int t  = threadIdx.x;                    // 0..255, owns j in [t*32, t*32+32)
  const int ti = tgt[i];
  const float* row = dist + (size_t)i * NTOT;

  unsigned ku[32];
  {
    const float4* p4 = (const float4*)(row + t * 32);
    #pragma unroll
    for (int q4 = 0; q4 < 8; ++q4) {
      float4 v = p4[q4];
      ku[4 * q4 + 0] = __float_as_uint(v.x);
      ku[4 * q4 + 1] = __float_as_uint(v.y);
      ku[4 * q4 + 2] = __float_as_uint(v.z);
      ku[4 * q4 + 3] = __float_as_uint(v.w);
    }
  }
  #pragma unroll
  for (int q = 0; q < 32; ++q)
    if (t * 32 + q == i) ku[q] = INF_BITS;       // exclude diagonal (dm = inf)

  // Binary search the (K+1)-th smallest bit pattern (exact order statistic).
  unsigned lo = 0u, hi = INF_BITS;
  while (lo < hi) {
    const unsigned mid = lo + ((hi - lo) >> 1);
    int cnt = 0;
    #pragma unroll
    for (int q = 0; q < 32; ++q) cnt += (ku[q] <= mid) ? 1 : 0;
    cnt = blockSumI(cnt, si);                    // uniform control flow
    if (cnt >= KSEL + 1) hi = mid; else lo = mid + 1;
  }
  const float thr = __uint_as_float(hi);

  // Masked sums / counts / first-positive / any-negative.
  float ps = 0.f, ns = 0.f;
  int pc = 0, nc = 0, fp = 0x7FFFFFFF, an = 0;
  int tl[32];
  {
    const int4* p4 = (const int4*)(tgt + t * 32);
    #pragma unroll
    for (int q4 = 0; q4 < 8; ++q4) {
      int4 v = p4[q4];
      tl[4 * q4 + 0] = v.x; tl[4 * q4 + 1] = v.y;
      tl[4 * q4 + 2] = v.z; tl[4 * q4 + 3] = v.w;
    }
  }
  #pragma unroll
  for (int q = 0; q < 32; ++q) {
    const int  j    = t * 32 + q;
    const bool diag = (j == i);
    const bool same = (tl[q] == ti) && !diag;
    const float d   = __uint_as_float(ku[q]);    // +inf on diagonal
    const bool below = d < thr;
    const float ed   = below ? expf(-ALPHA_F * d) : 0.f;
    if (same)       { ps += ed; pc += below ? 1 : 0; fp = j < fp ? j : fp; }
    else if (!diag) { ns += ed; nc += below ? 1 : 0; an = 1; }
  }
  ps = blockSumF(ps, sf);
  ns = blockSumF(ns, sf);
  pc = blockSumI(pc, si);
  nc = blockSumI(nc, si);
  fp = blockMinI(fp, si);
  an = blockMaxI(an, si);

  if (t == 0) {
    const bool anyPos = (fp != 0x7FFFFFFF);
    const bool valid  = anyPos && (an != 0);
    float pos_eff = ps;
    if (pc == 0) pos_eff = anyPos ? expf(-ALPHA_F * row[fp]) : 0.f;  // fallback pair
    if (valid) {
      atomicAdd(&accum[0], -logf(pos_eff / (pos_eff + ns)));  // loss sum
      const int pca = (pc == 0) ? 1 : pc;
      if (pca > nc) atomicAdd(&accum[1], 1.f);                // accuracy count
      atomicAdd(&accum[2], (float)pc);                        // tp sum
      atomicAdd(&accum[3], (float)nc);                        // tn sum
    }
  }
}

// ---------------- K4: finalize -----------------------------------------------
__global__ void finalize_kernel(const float* __restrict__ accum, float* __restrict__ out) {
  const int t = threadIdx.x;
  if (t < 4) out[t] = accum[t] * (1.f / (float)NTOT);
}

extern "C" void kernel_launch(void* const* d_in, const int* in_sizes, int n_in,
                              void* d_out, int out_size, void* d_ws, size_t ws_size,
                              hipStream_t stream) {
  (void)in_sizes; (void)n_in; (void)out_size; (void)ws_size;
  const float* X   = (const float*)d_in[0];   // [N, D] fp32
  const int*   tgt = (const int*)d_in[1];     // [N] int
  float*       out = (float*)d_out;           // [4]: loss, accuracy, tp, tn

  // Workspace layout: sq[N] | accum[4..255 pad] | dist[N*N]
  float* sq    = (float*)d_ws;
  float* accum = sq + NTOT;
  float* dist  = accum + 256;                 // 16B-aligned (offset 33792 B)

  sqnorm_kernel<<<NTOT / 8, 256, 0, stream>>>(X, sq, accum);
  dim3 g2(NTOT / 64, NTOT / 128);             // x: 64-col blocks, y: 128-row blocks
  dist_kernel<<<g2, 256, 0, stream>>>(X, sq, dist);
  row_kernel<<<NTOT, 256, 0, stream>>>(dist, tgt, accum);
  finalize_kernel<<<1, 64, 0, stream>>>(accum, out);
}